// SparseGraphConv_32444182954024
// MI455X (gfx1250) — compile-verified
//
#include <hip/hip_runtime.h>
#include <hip/hip_bf16.h>

// ---------------------------------------------------------------------------
// SparseGraphConv on MI455X (gfx1250):
//   1) edge scatter: s[src] += X[dst] (fp32 global atomics), c[src] += 1
//   2) fused: agg = s/(c+1e-8); h = X@Wself^T + agg@Wnbr^T via
//      V_WMMA_F32_16X16X4_F32; LayerNorm epilogue in-block.
// ---------------------------------------------------------------------------

typedef float v2f __attribute__((ext_vector_type(2)));
typedef float v8f __attribute__((ext_vector_type(8)));

#define D 128          // feature dim (fixed by reference)
#define TILE_M 16      // rows per block
#define LDSPAD 4       // avoid LDS bank conflicts (row stride 132*4B, 16B aligned)

__global__ __launch_bounds__(256)
void edge_scatter_kernel(const float* __restrict__ X,
                         const int*   __restrict__ nbr_src,
                         const int*   __restrict__ nbr_dst,
                         float* __restrict__ s_acc,   // [N, D]
                         float* __restrict__ c_acc,   // [N]
                         int E) {
    const int gtid = blockIdx.x * blockDim.x + threadIdx.x;
    const int e    = gtid >> 5;          // one wave32 per edge
    const int lane = gtid & 31;          // lane covers 4 consecutive features
    if (e >= E) return;

    const int dst = nbr_dst[e];          // gather source row (uniform in wave)
    const int src = nbr_src[e];          // scatter target row (uniform in wave)

    const float4 v = ((const float4*)(X + (size_t)dst * D))[lane];
    float* out = s_acc + (size_t)src * D + lane * 4;
    // fp32 global atomics: lowers to global_atomic_add_f32 (no CAS loop)
    unsafeAtomicAdd(out + 0, v.x);
    unsafeAtomicAdd(out + 1, v.y);
    unsafeAtomicAdd(out + 2, v.z);
    unsafeAtomicAdd(out + 3, v.w);
    if (lane == 0) unsafeAtomicAdd(c_acc + src, 1.0f);
}

__global__ __launch_bounds__(256)
void fused_gemm_ln_kernel(const float* __restrict__ X,       // [N, D]
                          const float* __restrict__ W_self,  // [D, D] row-major [out][in]
                          const float* __restrict__ W_nbr,   // [D, D]
                          const float* __restrict__ gamma,   // [D]
                          const float* __restrict__ beta,    // [D]
                          const float* __restrict__ s_acc,   // [N, D]
                          const float* __restrict__ c_acc,   // [N]
                          float* __restrict__ out,           // [N, D]
                          int N) {
    __shared__ float Xs [TILE_M][D + LDSPAD];
    __shared__ float Ags[TILE_M][D + LDSPAD];
    __shared__ float Hs [TILE_M][D + LDSPAD];

    const int tid     = threadIdx.x;
    const int wave    = tid >> 5;        // 8 waves: wave w owns output cols [16w, 16w+16)
    const int lane    = tid & 31;
    const int half    = lane >> 4;       // K sub-offset selector (0 -> K+0/1, 1 -> K+2/3)
    const int l16     = lane & 15;
    const int rowBase = blockIdx.x * TILE_M;

    // ---- stage X tile and agg tile (scaled by 1/(count+1e-8)) into LDS ----
    for (int i = tid; i < TILE_M * (D / 4); i += 256) {
        const int r = i >> 5;            // row in tile
        const int q = i & 31;            // float4 index within row
        int g = rowBase + r; if (g >= N) g = N - 1;
        const float4 xv = ((const float4*)(X + (size_t)g * D))[q];
        *(float4*)&Xs[r][q * 4] = xv;
        const float inv = 1.0f / (c_acc[g] + 1e-8f);
        float4 sv = ((const float4*)(s_acc + (size_t)g * D))[q];
        sv.x *= inv; sv.y *= inv; sv.z *= inv; sv.w *= inv;
        *(float4*)&Ags[r][q * 4] = sv;
    }
    __syncthreads();

    // ---- dual GEMM: acc = Xs @ Wself^T + Ags @ Wnbr^T, one 16x16 tile/wave ----
    // A frag (16x4 f32): lanes 0-15 hold (M=l16, K=k0+0/1), lanes 16-31 (M=l16, K=k0+2/3)
    // B frag (4x16 f32): lanes 0-15 hold (K=k0+0/1, N=l16), lanes 16-31 (K=k0+2/3, N=l16)
    // B[k][n] = W[n][k] since y = x @ W^T.
    const int ncol = wave * 16 + l16;                    // this lane's output column
    const float* wsp = W_self + (size_t)ncol * D;
    const float* wnp = W_nbr  + (size_t)ncol * D;

    v8f acc = {};
    #pragma unroll
    for (int k0 = 0; k0 < D; k0 += 4) {
        const int kk = k0 + half * 2;
        v2f a, b;
        a.x = Xs[l16][kk];  a.y = Xs[l16][kk + 1];
        b.x = wsp[kk];      b.y = wsp[kk + 1];
        acc = __builtin_amdgcn_wmma_f32_16x16x4_f32(
            false, a, false, b, (short)0, acc, false, false);
    }
    #pragma unroll
    for (int k0 = 0; k0 < D; k0 += 4) {
        const int kk = k0 + half * 2;
        v2f a, b;
        a.x = Ags[l16][kk]; a.y = Ags[l16][kk + 1];
        b.x = wnp[kk];      b.y = wnp[kk + 1];
        acc = __builtin_amdgcn_wmma_f32_16x16x4_f32(
            false, a, false, b, (short)0, acc, false, false);
    }

    // ---- spill h tile to LDS: VGPR r holds (M = half*8 + r, N = l16) ----
    #pragma unroll
    for (int r = 0; r < 8; ++r) {
        Hs[half * 8 + r][wave * 16 + l16] = acc[r];
    }
    __syncthreads();

    // ---- LayerNorm epilogue: wave w handles rows 2w and 2w+1 ----
    const float4 g4 = ((const float4*)gamma)[lane];
    const float4 b4 = ((const float4*)beta)[lane];
    #pragma unroll
    for (int rr = 0; rr < 2; ++rr) {
        const int r = wave * 2 + rr;
        const int grow = rowBase + r;
        const float4 h4 = *(const float4*)&Hs[r][lane * 4];
        float sum = h4.x + h4.y + h4.z + h4.w;
        float sq  = h4.x * h4.x + h4.y * h4.y + h4.z * h4.z + h4.w * h4.w;
        #pragma unroll
        for (int off = 16; off >= 1; off >>= 1) {
            sum += __shfl_xor(sum, off, 32);
            sq  += __shfl_xor(sq,  off, 32);
        }
        const float mean = sum * (1.0f / D);
        const float var  = sq * (1.0f / D) - mean * mean;
        const float rstd = rsqrtf(var + 1e-5f);
        if (grow < N) {
            float4 o;
            o.x = (h4.x - mean) * rstd * g4.x + b4.x;
            o.y = (h4.y - mean) * rstd * g4.y + b4.y;
            o.z = (h4.z - mean) * rstd * g4.z + b4.z;
            o.w = (h4.w - mean) * rstd * g4.w + b4.w;
            ((float4*)(out + (size_t)grow * D))[lane] = o;
        }
    }
}

extern "C" void kernel_launch(void* const* d_in, const int* in_sizes, int n_in,
                              void* d_out, int out_size, void* d_ws, size_t ws_size,
                              hipStream_t stream) {
    const float* X       = (const float*)d_in[0];
    const float* W_self  = (const float*)d_in[1];
    const float* W_nbr   = (const float*)d_in[2];
    const float* gamma   = (const float*)d_in[3];
    const float* beta    = (const float*)d_in[4];
    const int*   nbr_src = (const int*)d_in[5];
    const int*   nbr_dst = (const int*)d_in[6];

    const int N = in_sizes[0] / D;
    const int E = in_sizes[5];

    float* s_acc = (float*)d_ws;                 // [N, D]
    float* c_acc = s_acc + (size_t)N * D;        // [N]

    // zero the accumulators (graph-capturable)
    hipMemsetAsync(d_ws, 0, ((size_t)N * D + N) * sizeof(float), stream);

    // edge scatter: one wave32 per edge
    {
        const long long total = (long long)E * 32;
        const int blocks = (int)((total + 255) / 256);
        edge_scatter_kernel<<<blocks, 256, 0, stream>>>(
            X, nbr_src, nbr_dst, s_acc, c_acc, E);
    }

    // fused scale + dual WMMA GEMM + LayerNorm: 16 rows per block
    {
        const int blocks = (N + TILE_M - 1) / TILE_M;
        fused_gemm_ln_kernel<<<blocks, 256, 0, stream>>>(
            X, W_self, W_nbr, gamma, beta, s_acc, c_acc, (float*)d_out, N);
    }
}